// RoI2Det_24421184045578
// MI455X (gfx1250) — compile-verified
//
#include <hip/hip_runtime.h>
#include <hip/hip_bf16.h>
#include <stdint.h>

// ---------------- problem constants (match reference) ----------------
#define BIMG   4
#define NROIS  4096
#define NCLS   80          // foreground classes (scores have 81 cols)
#define NCOL   81
#define IMG_W  1333.0f
#define IMG_H  800.0f
#define SCORE_THR 0.05f
#define IOU_THR   0.5f
#define MAX_PER_IMG 100
#define K_PRE  2000
#define CAP    4096        // gather capacity (>= K_PRE + histogram-bucket slack)
#define MAXR   4.135166556742356f   // |log(16/1000)|
#define NTILE  (K_PRE / 16)         // 125
#define NTILES (NTILE * NTILE)      // 15625

// histogram of float-bit keys, top 18 bits; scores in (0.05, 1.0)
#define NB    2304
#define BOFF  62771u       // 0x3D4CCCCD >> 14

// ---------------- workspace layout (bytes) ----------------
static constexpr size_t OFF_CUT  = 0;                                   // B u32
static constexpr size_t OFF_CNT  = 64;                                  // B u32
static constexpr size_t OFF_HIST = 256;                                 // B*NB u32 (36864)
static constexpr size_t OFF_CAND = OFF_HIST + (size_t)BIMG * NB * 4;    // B*CAP uint2 (131072)
static constexpr size_t OFF_BOX  = OFF_CAND + (size_t)BIMG * CAP * 8;   // B*K_PRE float4
static constexpr size_t OFF_SC   = OFF_BOX  + (size_t)BIMG * K_PRE * 16;
static constexpr size_t OFF_CLS  = OFF_SC   + (size_t)BIMG * K_PRE * 4;
static constexpr size_t OFF_MASK = OFF_CLS  + (size_t)BIMG * K_PRE * 4; // B*K_PRE*128 u16

typedef float v2f __attribute__((ext_vector_type(2)));
typedef float v8f __attribute__((ext_vector_type(8)));
typedef int   agv4i __attribute__((vector_size(16)));   // matches builtin's V4i param

// ---------------- stage 0: zero per-image global histogram + counters ----------------
__global__ __launch_bounds__(256) void k_init(unsigned* __restrict__ hist,
                                              unsigned* __restrict__ cnt) {
  const int b = blockIdx.x, tid = threadIdx.x;
  for (int i = tid; i < NB; i += 256) hist[(size_t)b * NB + i] = 0u;
  if (tid == 0) cnt[b] = 0u;
}

// ---------------- stage 1: softmax + histogram (16 blocks/image) ----------------
__global__ __launch_bounds__(256) void k_hist(const float* __restrict__ clsout,
                                              unsigned* __restrict__ hist) {
  const int b = blockIdx.y, tid = threadIdx.x;
  const int r = blockIdx.x * 256 + tid;              // one row per thread
  __shared__ unsigned h[NB];
  for (int i = tid; i < NB; i += 256) h[i] = 0u;
  __syncthreads();
  const float* p = clsout + ((size_t)b * NROIS + r) * NCOL;
  __builtin_prefetch(p + 256 * NCOL, 0, 1);          // global_prefetch_b8
  float m = p[0];
  for (int c = 1; c < NCOL; ++c) m = fmaxf(m, p[c]);
  float s = 0.f;
  for (int c = 0; c < NCOL; ++c) s += expf(p[c] - m);
  const float inv = 1.0f / s;
  for (int c = 0; c < NCLS; ++c) {
    const float sc = expf(p[c] - m) * inv;
    if (sc > SCORE_THR) {
      const unsigned key = __float_as_uint(sc);
      int bk = (int)(key >> 14) - (int)BOFF;
      bk = bk < 0 ? 0 : (bk >= NB ? NB - 1 : bk);
      atomicAdd(&h[bk], 1u);
    }
  }
  __syncthreads();
  for (int i = tid; i < NB; i += 256) {
    const unsigned v = h[i];
    if (v) atomicAdd(&hist[(size_t)b * NB + i], v);
  }
}

// ---------------- stage 1b: scan histogram -> per-image cut key ----------------
__global__ __launch_bounds__(32) void k_cut(const unsigned* __restrict__ hist,
                                            unsigned* __restrict__ cut) {
  const int b = blockIdx.x;
  if (threadIdx.x == 0) {
    const unsigned* h = hist + (size_t)b * NB;
    unsigned cum = 0; int cb = 0;
    for (int i = NB - 1; i >= 0; --i) { cum += h[i]; if (cum >= K_PRE) { cb = i; break; } }
    cut[b] = ((unsigned)cb + BOFF) << 14;
  }
}

// ---------------- stage 2: gather candidates with key >= cut (16 blocks/image) ----------------
__global__ __launch_bounds__(256) void k_gather(const float* __restrict__ clsout,
                                                const unsigned* __restrict__ cut,
                                                unsigned* __restrict__ cnt,
                                                uint2* __restrict__ cand) {
  const int b = blockIdx.y, tid = threadIdx.x;
  const int r = blockIdx.x * 256 + tid;
  const unsigned cutLo = cut[b];
  const float* p = clsout + ((size_t)b * NROIS + r) * NCOL;
  __builtin_prefetch(p + 256 * NCOL, 0, 1);
  float m = p[0];
  for (int c = 1; c < NCOL; ++c) m = fmaxf(m, p[c]);
  float s = 0.f;
  for (int c = 0; c < NCOL; ++c) s += expf(p[c] - m);
  const float inv = 1.0f / s;
  for (int c = 0; c < NCLS; ++c) {
    const float sc = expf(p[c] - m) * inv;
    if (sc > SCORE_THR) {
      const unsigned key = __float_as_uint(sc);
      if (key >= cutLo) {
        const unsigned pos = atomicAdd(&cnt[b], 1u);
        if (pos < CAP) cand[(size_t)b * CAP + pos] = make_uint2(key, (unsigned)(r * NCLS + c));
      }
    }
  }
}

// ---------------- stage 3: bitonic sort (score desc, idx asc), keep top K_PRE ----------------
__global__ __launch_bounds__(256) void k_sort(uint2* __restrict__ cand,
                                              const unsigned* __restrict__ cnt) {
  const int b = blockIdx.x, tid = threadIdx.x;
  __shared__ unsigned long long sm[CAP];   // 32 KB
  unsigned n = cnt[b]; if (n > CAP) n = CAP;
  const uint2* src = cand + (size_t)b * CAP;

#if __has_builtin(__builtin_amdgcn_global_load_async_to_lds_b128)
  // gfx1250 async copy: global -> LDS, tracked by ASYNCcnt (16B = 2 pairs per op)
  for (int i = tid * 2; i < CAP; i += 256 * 2) {
    __builtin_amdgcn_global_load_async_to_lds_b128(
        (__attribute__((address_space(1))) agv4i*)(src + i),
        (__attribute__((address_space(3))) agv4i*)(sm + i), 0, 0);
  }
#if __has_builtin(__builtin_amdgcn_s_wait_asynccnt)
  __builtin_amdgcn_s_wait_asynccnt(0);
#else
  asm volatile("s_wait_asynccnt 0" ::: "memory");
#endif
  __syncthreads();
  for (int i = tid; i < CAP; i += 256) {
    unsigned long long v = 0ull;
    if ((unsigned)i < n) {
      const uint2 p = ((const uint2*)sm)[i];          // in-place reinterpret
      v = ((unsigned long long)p.x << 32) | (unsigned long long)(0xFFFFFFFFu - p.y);
    }
    sm[i] = v;
  }
#else
  for (int i = tid; i < CAP; i += 256) {
    unsigned long long v = 0ull;
    if ((unsigned)i < n) {
      const uint2 p = src[i];
      v = ((unsigned long long)p.x << 32) | (unsigned long long)(0xFFFFFFFFu - p.y);
    }
    sm[i] = v;
  }
#endif
  __syncthreads();
  for (int k = 2; k <= CAP; k <<= 1) {
    for (int j = k >> 1; j > 0; j >>= 1) {
      for (int i = tid; i < CAP; i += 256) {
        const int ixj = i ^ j;
        if (ixj > i) {
          const unsigned long long A = sm[i], B = sm[ixj];
          const bool sw = ((i & k) == 0) ? (A < B) : (A > B);  // descending
          if (sw) { sm[i] = B; sm[ixj] = A; }
        }
      }
      __syncthreads();
    }
  }
  for (int i = tid; i < K_PRE; i += 256) {
    const unsigned long long v = sm[i];
    const unsigned key = (unsigned)(v >> 32);
    const unsigned idx = key ? (0xFFFFFFFFu - (unsigned)(v & 0xFFFFFFFFull)) : 0u;
    cand[(size_t)b * CAP + i] = make_uint2(key, idx);
  }
}

// ---------------- stage 4: decode top-K_PRE boxes ----------------
__global__ __launch_bounds__(256) void k_decode(const float* __restrict__ reg,
                                                const float* __restrict__ rois,
                                                const uint2* __restrict__ cand,
                                                float* __restrict__ box,
                                                float* __restrict__ sc,
                                                int* __restrict__ cls) {
  const int b = blockIdx.y;
  const int k = blockIdx.x * 256 + threadIdx.x;
  if (k >= K_PRE) return;
  const uint2 p = cand[(size_t)b * CAP + k];
  const float s = __uint_as_float(p.x);
  const unsigned flat = p.y;
  const unsigned row = flat / NCLS, cl = flat % NCLS;
  const float4 r4 = *(const float4*)(rois + ((size_t)b * NROIS + row) * 4);
  const float4 d4 = *(const float4*)(reg + ((size_t)b * NROIS + row) * (4 * NCLS) + cl * 4);
  const float px = (r4.x + r4.z) * 0.5f, py = (r4.y + r4.w) * 0.5f;
  const float pw = r4.z - r4.x, ph = r4.w - r4.y;
  const float dx = d4.x * 0.1f, dy = d4.y * 0.1f;
  const float dw = fminf(d4.z * 0.2f, MAXR), dh = fminf(d4.w * 0.2f, MAXR);
  const float gx = px + pw * dx, gy = py + ph * dy;
  const float gw = pw * expf(dw), gh = ph * expf(dh);
  float x1 = gx - gw * 0.5f, y1 = gy - gh * 0.5f, x2 = gx + gw * 0.5f, y2 = gy + gh * 0.5f;
  x1 = fminf(fmaxf(x1, 0.f), IMG_W); x2 = fminf(fmaxf(x2, 0.f), IMG_W);
  y1 = fminf(fmaxf(y1, 0.f), IMG_H); y2 = fminf(fmaxf(y2, 0.f), IMG_H);
  ((float4*)box)[(size_t)b * K_PRE + k] = make_float4(x1, y1, x2, y2);
  sc[(size_t)b * K_PRE + k] = s;
  cls[(size_t)b * K_PRE + k] = (int)cl;
}

// ---------------- stage 5: IoU suppression mask, 16x16 tiles via WMMA ----------------
// area_i + area_j over a tile is a rank-2 matmul: A=[area,1,0,0] (16x4), B=[1;area;0;0] (4x16)
// 4 waves per workgroup; tile index is wave-uniform so EXEC stays all-ones for WMMA.
__global__ __launch_bounds__(128) void k_mask(const float* __restrict__ box,
                                              const int* __restrict__ cls,
                                              unsigned short* __restrict__ mask) {
  const int b = blockIdx.y;
  const int wave = threadIdx.x >> 5;
  const int lane = threadIdx.x & 31;
  const int tile = blockIdx.x * 4 + wave;
  if (tile >= NTILES) return;
  const int ti = tile / NTILE, tj = tile % NTILE;
  const int rowBase = ti * 16, colBase = tj * 16;
  const float4* Bx = (const float4*)box + (size_t)b * K_PRE;
  const int* Cl = cls + (size_t)b * K_PRE;

  v2f a = {0.f, 0.f}, bv = {0.f, 0.f};
  if (lane < 16) {
    const float4 rb = Bx[rowBase + lane];
    a.x = (rb.z - rb.x) * (rb.w - rb.y);  // area of row box (K=0)
    a.y = 1.0f;                           // K=1
    const float4 cb = Bx[colBase + lane];
    bv.x = 1.0f;                          // row K=0 of B
    bv.y = (cb.z - cb.x) * (cb.w - cb.y); // row K=1 of B
  }
  v8f c = {0.f, 0.f, 0.f, 0.f, 0.f, 0.f, 0.f, 0.f};
  // D[m][n] = area_row[m] + area_col[n]  (exact f32, matches reference)
  v8f d = __builtin_amdgcn_wmma_f32_16x16x4_f32(false, a, false, bv,
                                                (short)0, c, false, false);

  const int half = lane >> 4;       // 0: rows r, 1: rows r+8
  const int n = lane & 15;
  const int j = colBase + n;
  const float4 bj = Bx[j];
  const int cj = Cl[j];
  for (int r = 0; r < 8; ++r) {
    const int i = rowBase + r + half * 8;
    const float4 bi = Bx[i];
    const int ci = Cl[i];
    const float S = d[r];
    const float iw = fminf(bi.z, bj.z) - fmaxf(bi.x, bj.x);
    const float ih = fminf(bi.w, bj.w) - fmaxf(bi.y, bj.y);
    const float inter = fmaxf(iw, 0.f) * fmaxf(ih, 0.f);
    const float denom = fmaxf(S - inter, 1e-6f);
    const bool pred = (j > i) && (ci == cj) && ((inter / denom) > IOU_THR);
    const unsigned bal = (unsigned)__ballot(pred);    // wave32 mask
    if (lane == 0) {
      const size_t rb0 = ((size_t)b * K_PRE + rowBase + r) * 128 + tj;
      const size_t rb1 = ((size_t)b * K_PRE + rowBase + r + 8) * 128 + tj;
      mask[rb0] = (unsigned short)(bal & 0xFFFFu);
      mask[rb1] = (unsigned short)(bal >> 16);
    }
  }
}

// ---------------- stage 6: serial greedy scan + final top-100 output ----------------
__global__ __launch_bounds__(128) void k_scan_out(const float* __restrict__ box,
                                                  const float* __restrict__ sc,
                                                  const int* __restrict__ cls,
                                                  const unsigned short* __restrict__ mask,
                                                  float* __restrict__ out) {
  const int b = blockIdx.x, tid = threadIdx.x;
  __shared__ float sc_sh[K_PRE];
  __shared__ unsigned short sup[128];
  for (int i = tid; i < K_PRE; i += 128) sc_sh[i] = sc[(size_t)b * K_PRE + i];
  if (tid < 128) sup[tid] = 0;
  __syncthreads();
  for (int i = 0; i < K_PRE; ++i) {
    const bool bit = (sup[i >> 4] >> (i & 15)) & 1;
    const bool active = (sc_sh[i] > 0.f) && !bit;
    __syncthreads();
    if (active && tid < NTILE) {
      sup[tid] |= mask[((size_t)b * K_PRE + i) * 128 + tid];
    }
    __syncthreads();
  }
  if (tid == 0) {
    float* det_b = out;                               // [B,100,4]
    float* det_s = out + (size_t)BIMG * MAX_PER_IMG * 4;
    float* det_l = det_s + (size_t)BIMG * MAX_PER_IMG;
    int k = 0;
    for (int i = 0; i < K_PRE && k < MAX_PER_IMG; ++i) {
      const bool bit = (sup[i >> 4] >> (i & 15)) & 1;
      const float s = sc_sh[i];
      if (s > 0.f && !bit) {
        const size_t src = ((size_t)b * K_PRE + i) * 4;
        const size_t dst = ((size_t)b * MAX_PER_IMG + k) * 4;
        det_b[dst + 0] = box[src + 0];
        det_b[dst + 1] = box[src + 1];
        det_b[dst + 2] = box[src + 2];
        det_b[dst + 3] = box[src + 3];
        det_s[(size_t)b * MAX_PER_IMG + k] = s;
        det_l[(size_t)b * MAX_PER_IMG + k] = (float)cls[(size_t)b * K_PRE + i];
        ++k;
      }
    }
    for (; k < MAX_PER_IMG; ++k) {
      const size_t dst = ((size_t)b * MAX_PER_IMG + k) * 4;
      det_b[dst + 0] = 0.f; det_b[dst + 1] = 0.f; det_b[dst + 2] = 0.f; det_b[dst + 3] = 0.f;
      det_s[(size_t)b * MAX_PER_IMG + k] = 0.f;
      det_l[(size_t)b * MAX_PER_IMG + k] = -1.0f;
    }
  }
}

// ---------------- host launcher ----------------
extern "C" void kernel_launch(void* const* d_in, const int* in_sizes, int n_in,
                              void* d_out, int out_size, void* d_ws, size_t ws_size,
                              hipStream_t stream) {
  (void)in_sizes; (void)n_in; (void)out_size; (void)ws_size;
  const float* clsout = (const float*)d_in[0];   // [B,N,81]
  const float* reg    = (const float*)d_in[1];   // [B,N,320]
  const float* rois   = (const float*)d_in[2];   // [B,N,4]

  uint8_t* ws = (uint8_t*)d_ws;
  unsigned*       cut  = (unsigned*)(ws + OFF_CUT);
  unsigned*       cnt  = (unsigned*)(ws + OFF_CNT);
  unsigned*       hist = (unsigned*)(ws + OFF_HIST);
  uint2*          cand = (uint2*)(ws + OFF_CAND);
  float*          box  = (float*)(ws + OFF_BOX);
  float*          sc   = (float*)(ws + OFF_SC);
  int*            cls  = (int*)(ws + OFF_CLS);
  unsigned short* mask = (unsigned short*)(ws + OFF_MASK);
  float*          out  = (float*)d_out;

  k_init  <<<BIMG, 256, 0, stream>>>(hist, cnt);
  k_hist  <<<dim3(NROIS / 256, BIMG), 256, 0, stream>>>(clsout, hist);
  k_cut   <<<BIMG, 32, 0, stream>>>(hist, cut);
  k_gather<<<dim3(NROIS / 256, BIMG), 256, 0, stream>>>(clsout, cut, cnt, cand);
  k_sort  <<<BIMG, 256, 0, stream>>>(cand, cnt);
  k_decode<<<dim3((K_PRE + 255) / 256, BIMG), 256, 0, stream>>>(reg, rois, cand, box, sc, cls);
  k_mask  <<<dim3((NTILES + 3) / 4, BIMG), 128, 0, stream>>>(box, cls, mask);
  k_scan_out<<<BIMG, 128, 0, stream>>>(box, sc, cls, mask, out);
}